// AxialAttention_28750511079836
// MI455X (gfx1250) — compile-verified
//
#include <hip/hip_runtime.h>

typedef float v2f __attribute__((ext_vector_type(2)));
typedef float v8f __attribute__((ext_vector_type(8)));
typedef int   i32x4 __attribute__((vector_size(16)));

#define AS1 __attribute__((address_space(1)))
#define AS3 __attribute__((address_space(3)))

static constexpr int T   = 128;   // sequence length (axial dim)
static constexpr int Dm  = 128;   // embedding dim
static constexpr int HE  = 16;    // head dim (D / 8 heads)
static constexpr int LDA = 132;   // padded LDS row stride (floats): 16B-aligned, bank-spread

__device__ __forceinline__ v8f wmma4(v2f a, v2f b, v8f c) {
  // D = A(16x4 f32) * B(4x16 f32) + C(16x16 f32)
  return __builtin_amdgcn_wmma_f32_16x16x4_f32(false, a, false, b, (short)0, c,
                                               false, false);
}

__global__ __launch_bounds__(256, 1)
void axial_pass_kernel(const float* __restrict__ x,
                       const float* __restrict__ Wq,
                       const float* __restrict__ Wkv,
                       const float* __restrict__ Wo,
                       const float* __restrict__ bo,
                       float* __restrict__ out,
                       int pass)
{
  __shared__ float xs[T * LDA];   // x_seq, later per-wave softmax scratch
  __shared__ float qs[T * LDA];   // Q, later attention output
  __shared__ float ks[T * LDA];
  __shared__ float vs[T * LDA];

  const int tid  = threadIdx.x;
  const int wave = tid >> 5;         // 8 waves; also head id; also N-strip id
  const int lane = tid & 31;
  const int nl   = lane & 15;        // N / M index within a tile
  const int kh   = (lane >> 4) << 1; // K-half offset: 0 or 2
  const int rh   = (lane >> 4) << 3; // C row-half offset: 0 or 8

  // ---- sequence addressing ------------------------------------------------
  const int s = blockIdx.x;
  const int b = s >> 7, o = s & 127;
  int base, tstride;
  if (pass == 0) {              // attend along H: x[b, t, o, :]
    base    = b * (128 * 128 * 128) + o * 128;
    tstride = 128 * 128;
  } else {                      // attend along W: x[b, o, t, :]
    base    = (b * 128 + o) * (128 * 128);
    tstride = 128;
  }

  // ---- stage x_seq [128 x 128] into LDS (async-to-LDS path on CDNA5) ------
#if __has_builtin(__builtin_amdgcn_global_load_async_to_lds_b128)
  for (int i = tid; i < T * (Dm / 4); i += 256) {
    const int row = i >> 5;
    const int c4  = (i & 31) << 2;
    __builtin_amdgcn_global_load_async_to_lds_b128(
        (AS1 i32x4*)(x + base + row * tstride + c4),
        (AS3 i32x4*)&xs[row * LDA + c4], 0, 0);
  }
#if __has_builtin(__builtin_amdgcn_s_wait_asynccnt)
  __builtin_amdgcn_s_wait_asynccnt(0);
#else
  asm volatile("s_wait_asynccnt 0" ::: "memory");
#endif
#else
  for (int i = tid; i < T * (Dm / 4); i += 256) {
    const int row = i >> 5;
    const int c4  = (i & 31) << 2;
    const float4 v = *(const float4*)(x + base + row * tstride + c4);
    *(float4*)&xs[row * LDA + c4] = v;
  }
#endif
  __syncthreads();

  // ---- column-strip GEMM: OutL[:, oc0..oc0+15] = A(128x128,LDS) @ Bg ------
  auto gemm_col = [&](const float* __restrict__ A, const float* __restrict__ Bg,
                      int ldb, int bc0, float* __restrict__ OutL, int oc0) {
    v8f c[8];
#pragma unroll
    for (int i = 0; i < 8; ++i) c[i] = (v8f){0.f,0.f,0.f,0.f,0.f,0.f,0.f,0.f};
#pragma unroll 2
    for (int k0 = 0; k0 < Dm; k0 += 4) {
      const int kk = k0 + kh;
      v2f bb;
      bb.x = Bg[(kk + 0) * ldb + bc0 + nl];
      bb.y = Bg[(kk + 1) * ldb + bc0 + nl];
#pragma unroll
      for (int mt = 0; mt < 8; ++mt) {
        const v2f aa = *(const v2f*)&A[(mt * 16 + nl) * LDA + kk];
        c[mt] = wmma4(aa, bb, c[mt]);
      }
    }
#pragma unroll
    for (int mt = 0; mt < 8; ++mt)
#pragma unroll
      for (int r = 0; r < 8; ++r)
        OutL[(mt * 16 + r + rh) * LDA + oc0 + nl] = c[mt][r];
  };

  // ---- projections: Q, K, V ----------------------------------------------
  gemm_col(xs, Wq,  Dm,     wave * 16,      qs, wave * 16);
  gemm_col(xs, Wkv, 2 * Dm, wave * 16,      ks, wave * 16);
  gemm_col(xs, Wkv, 2 * Dm, Dm + wave * 16, vs, wave * 16);
  __syncthreads();   // q/k/v ready; xs free for reuse as softmax scratch

  // ---- attention: one wave per head --------------------------------------
  const int h = wave;
  float* scratch = &xs[wave * 16 * LDA];   // per-wave 16x128 probability strip
  const float scale = 0.25f;               // 1/sqrt(16)

#pragma unroll 1
  for (int mt = 0; mt < 8; ++mt) {
    const int m0 = mt * 16;

    // S strip (16 x 128) = Q_h[m0:m0+16, :] @ K_h^T
    v8f sc[8];
#pragma unroll
    for (int i = 0; i < 8; ++i) sc[i] = (v8f){0.f,0.f,0.f,0.f,0.f,0.f,0.f,0.f};
#pragma unroll
    for (int nt = 0; nt < 8; ++nt)
#pragma unroll
      for (int k0 = 0; k0 < HE; k0 += 4) {
        const int kk = h * 16 + k0 + kh;
        const v2f aa = *(const v2f*)&qs[(m0 + nl) * LDA + kk];
        const v2f bb = *(const v2f*)&ks[(nt * 16 + nl) * LDA + kk];
        sc[nt] = wmma4(aa, bb, sc[nt]);
      }

    // softmax over each row (row = 16 lanes within a wave half x 8 N-tiles)
#pragma unroll
    for (int r = 0; r < 8; ++r) {
      float mx = -3.4e38f;
#pragma unroll
      for (int nt = 0; nt < 8; ++nt) mx = fmaxf(mx, sc[nt][r]);
#pragma unroll
      for (int off = 1; off < 16; off <<= 1) mx = fmaxf(mx, __shfl_xor(mx, off, 32));
      float p[8], sum = 0.f;
#pragma unroll
      for (int nt = 0; nt < 8; ++nt) {
        p[nt] = __expf((sc[nt][r] - mx) * scale);
        sum += p[nt];
      }
#pragma unroll
      for (int off = 1; off < 16; off <<= 1) sum += __shfl_xor(sum, off, 32);
      const float inv = 1.f / sum;
      const int row = r + rh;
#pragma unroll
      for (int nt = 0; nt < 8; ++nt)
        scratch[row * LDA + nt * 16 + nl] = p[nt] * inv;
    }

    // O strip (16 x 16) = P(16x128) @ V_h(128x16)
    v8f oa = (v8f){0.f,0.f,0.f,0.f,0.f,0.f,0.f,0.f};
#pragma unroll 2
    for (int k0 = 0; k0 < T; k0 += 4) {
      const int kk = k0 + kh;
      const v2f aa = *(const v2f*)&scratch[nl * LDA + kk];
      v2f bb;
      bb.x = vs[(kk + 0) * LDA + h * 16 + nl];
      bb.y = vs[(kk + 1) * LDA + h * 16 + nl];
      oa = wmma4(aa, bb, oa);
    }
#pragma unroll
    for (int r = 0; r < 8; ++r)
      qs[(m0 + r + rh) * LDA + h * 16 + nl] = oa[r];   // overwrite own head cols
  }
  __syncthreads();   // qs now holds the full attention output

  // ---- output projection + bias (+ accumulate for pass 1) -----------------
  {
    v8f c[8];
#pragma unroll
    for (int i = 0; i < 8; ++i) c[i] = (v8f){0.f,0.f,0.f,0.f,0.f,0.f,0.f,0.f};
#pragma unroll 2
    for (int k0 = 0; k0 < Dm; k0 += 4) {
      const int kk = k0 + kh;
      v2f bb;
      bb.x = Wo[(kk + 0) * Dm + wave * 16 + nl];
      bb.y = Wo[(kk + 1) * Dm + wave * 16 + nl];
#pragma unroll
      for (int mt = 0; mt < 8; ++mt) {
        const v2f aa = *(const v2f*)&qs[(mt * 16 + nl) * LDA + kk];
        c[mt] = wmma4(aa, bb, c[mt]);
      }
    }
    const float bias = bo[wave * 16 + nl];
#pragma unroll
    for (int mt = 0; mt < 8; ++mt)
#pragma unroll
      for (int r = 0; r < 8; ++r) {
        const int tok = mt * 16 + r + rh;
        const int off = base + tok * tstride + wave * 16 + nl;
        float v = c[mt][r] + bias;
        if (pass) v += out[off];
        out[off] = v;
      }
  }
}

extern "C" void kernel_launch(void* const* d_in, const int* in_sizes, int n_in,
                              void* d_out, int out_size, void* d_ws, size_t ws_size,
                              hipStream_t stream) {
  const float* x    = (const float*)d_in[0];
  const float* Wq0  = (const float*)d_in[1];
  const float* Wkv0 = (const float*)d_in[2];
  const float* Wo0  = (const float*)d_in[3];
  const float* bo0  = (const float*)d_in[4];
  const float* Wq1  = (const float*)d_in[5];
  const float* Wkv1 = (const float*)d_in[6];
  const float* Wo1  = (const float*)d_in[7];
  const float* bo1  = (const float*)d_in[8];
  float* out = (float*)d_out;

  dim3 grid(1024), block(256);
  // pass 0 (attend along H) writes out; pass 1 (attend along W) accumulates.
  axial_pass_kernel<<<grid, block, 0, stream>>>(x, Wq0, Wkv0, Wo0, bo0, out, 0);
  axial_pass_kernel<<<grid, block, 0, stream>>>(x, Wq1, Wkv1, Wo1, bo1, out, 1);
}